// OrthBILinear_16192026705944
// MI455X (gfx1250) — compile-verified
//
#include <hip/hip_runtime.h>
#include <hip/hip_bf16.h>

// Problem constants (match the reference).
#define GROUP 16
#define BS    8192
#define CH    256

typedef float v2f __attribute__((ext_vector_type(2)));
typedef float v8f __attribute__((ext_vector_type(8)));
typedef unsigned int u32x4 __attribute__((ext_vector_type(4)));
typedef int          i32x4 __attribute__((ext_vector_type(4)));
typedef int          i32x8 __attribute__((ext_vector_type(8)));

#if defined(__has_builtin)
#if __has_builtin(__builtin_amdgcn_tensor_load_to_lds) && \
    __has_builtin(__builtin_amdgcn_s_wait_tensorcnt)
#define HAVE_TDM 1
#endif
#endif
#ifndef HAVE_TDM
#define HAVE_TDM 0
#endif

// ---------------------------------------------------------------------------
// Generic batched f32 WMMA GEMM:  C = [I -] op(A) @ op(B)
//   TA / TB: logical transposes; IDENT_MINUS: C = I - acc; NT_STORE: streaming.
// One wave -> one 16x16 C tile, K-loop via v_wmma_f32_16x16x4_f32.
// Used for the three tiny 256^3 factor GEMMs (gram / V*T / Qt assembly).
// ---------------------------------------------------------------------------
template<bool TA, bool TB, bool IDENT_MINUS, bool NT_STORE>
__global__ __launch_bounds__(128)
void wmma_gemm_f32(const float* __restrict__ A, const float* __restrict__ B,
                   float* __restrict__ C,
                   int M, int N, int K,
                   long sA, long sB, long sC,
                   int lda, int ldb, int ldc)
{
    const int g = blockIdx.z;
    A += (long)g * sA;
    B += (long)g * sB;
    C += (long)g * sC;

    const int wave = threadIdx.x >> 5;
    const int lane = threadIdx.x & 31;
    const int half = lane >> 4;   // 0: lanes 0-15, 1: lanes 16-31
    const int l16  = lane & 15;

    const int m0 = (blockIdx.y * 4 + wave) * 16;
    const int n0 = blockIdx.x * 16;

    v8f acc = {};
    #pragma unroll 4
    for (int k = 0; k < K; k += 4) {
        const int kk = k + 2 * half;              // this lane-half's k base
        v2f a, b;
        if (!TA) {
            a = *(const v2f*)(A + (size_t)(m0 + l16) * lda + kk);
        } else {
            a.x = A[(size_t)kk       * lda + (m0 + l16)];
            a.y = A[(size_t)(kk + 1) * lda + (m0 + l16)];
        }
        if (!TB) {
            b.x = B[(size_t)kk       * ldb + (n0 + l16)];
            b.y = B[(size_t)(kk + 1) * ldb + (n0 + l16)];
        } else {
            b = *(const v2f*)(B + (size_t)(n0 + l16) * ldb + kk);
        }
        acc = __builtin_amdgcn_wmma_f32_16x16x4_f32(
                  false, a, false, b, (short)0, acc, false, false);
    }

    #pragma unroll
    for (int r = 0; r < 8; ++r) {
        const int row = m0 + r + 8 * half;        // C/D layout: M = r + 8*half
        const int col = n0 + l16;
        float v = acc[r];
        if (IDENT_MINUS) v = ((row == col) ? 1.0f : 0.0f) - v;
        float* p = C + (size_t)row * ldc + col;
        if (NT_STORE) __builtin_nontemporal_store(v, p);
        else          *p = v;
    }
}

// ---------------------------------------------------------------------------
// Forward-WY triangular factor (LAPACK larft):
//   T[i][i]  = tau_i = 2 / G[i][i]
//   T[0:i,i] = -tau_i * T[0:i,0:i] @ G[0:i,i]
// Thread j owns row j of T in LDS (stride 257 -> conflict-free; 257 KB fits
// the 320 KB WGP LDS). T[j][*] is only ever touched by thread j, and G is
// symmetric (read row i instead of column i -> coalesced), so the whole
// recurrence needs NO barriers.
// ---------------------------------------------------------------------------
__global__ __launch_bounds__(256)
void larft_f32(const float* __restrict__ G, float* __restrict__ T)
{
    extern __shared__ float Ts[];                 // 256 * 257 floats (dynamic)
    const int g = blockIdx.x;
    const int j = threadIdx.x;
    const float* __restrict__ Gg = G + (size_t)g * CH * CH;
    float*       __restrict__ Tg = T + (size_t)g * CH * CH;

    const int LD = CH + 1;
    float* row = Ts + (size_t)j * LD;

    #pragma unroll 8
    for (int k = 0; k < CH; ++k) row[k] = 0.0f;

    for (int i = 0; i < CH; ++i) {
        const float taui = 2.0f / Gg[(size_t)i * CH + i];
        if (j < i) {
            const float* __restrict__ gi = Gg + (size_t)i * CH;
            float s = 0.0f;
            for (int k = j; k < i; ++k) s += row[k] * gi[k];
            row[i] = -taui * s;
        } else if (j == i) {
            row[i] = taui;
        }
    }

    #pragma unroll 8
    for (int k = 0; k < CH; ++k) Tg[(size_t)j * CH + k] = row[k];
}

// ---------------------------------------------------------------------------
// Big streaming GEMM:  out[b][c] = sum_k x[b][k] * Qt[c][k]
// Block = 256 threads = 8 waves, all sharing one 16-column slab of Qt
// (16 rows x 256 k, f32). The slab is pulled into LDS once per block by the
// Tensor Data Mover (tensor_load_to_lds) with pad_enable: +4 dwords of pad
// per 256-dword row => LDS row stride 260 => B-fragment ds_load_b64 banks are
// (4c + k) mod 64: conflict-free across the wave. Wave 0 issues the TDM and
// waits on TENSORcnt; everyone syncs on the workgroup barrier.
// ---------------------------------------------------------------------------
#define SLAB_LD 260                                // 256 + 4 pad dwords

__global__ __launch_bounds__(256)
void xq_gemm_tdm(const float* __restrict__ x, const float* __restrict__ Qt,
                 float* __restrict__ out)
{
    extern __shared__ float slab[];                // 16 * 260 dwords = 16,640 B

    const int g  = blockIdx.z;
    const int n0 = blockIdx.x * 16;

    const float* __restrict__ xg = x   + (size_t)g * BS * CH;
    float*       __restrict__ og = out + (size_t)g * BS * CH;
    const float* __restrict__ qg = Qt  + (size_t)g * CH * CH + (size_t)n0 * CH;

    const int wave = threadIdx.x >> 5;
    const int lane = threadIdx.x & 31;
    const int half = lane >> 4;
    const int l16  = lane & 15;

#if HAVE_TDM
    if (wave == 0) {
        // D# per ISA 8.3/8.4: 2D f32 tile, 256 wide x 16 rows, padded.
        const unsigned long long ga = (unsigned long long)(uintptr_t)qg;
        u32x4 d0;
        d0.x = 1u;                                   // count=1, user mode
        d0.y = 0u;                                   // lds_addr = 0 (dyn LDS base)
        d0.z = (unsigned int)(ga & 0xFFFFFFFFu);     // global_addr lo
        d0.w = (unsigned int)(ga >> 32) | (2u << 30);// global_addr hi | type=2
        i32x8 d1;
        d1[0] = (int)((2u << 16)      // data_size = 4 bytes
                    | (1u << 20)      // pad_enable
                    | (7u << 22)      // pad_interval: every 256 dwords
                    | (3u << 25));    // pad_amount: 4 dwords
        d1[1] = (int)(256u << 16);    // tensor_dim0 = 256
        d1[2] = (int)(256u << 16);    // tensor_dim1 = 256
        d1[3] = (int)(256u << 16);    // tile_dim0 = 256
        d1[4] = 16;                   // tile_dim1 = 16
        d1[5] = 256;                  // tensor_dim0_stride = 256
        d1[6] = 0;
        d1[7] = 0;
        i32x4 dz4 = {0, 0, 0, 0};     // <=2D: groups 2/3 unused
        i32x8 dz8 = {0, 0, 0, 0, 0, 0, 0, 0};
        // clang-23 / therock-10.0 arity: (g0, g1, g2, g3, extra, cpol)
        __builtin_amdgcn_tensor_load_to_lds(d0, d1, dz4, dz4, dz8, 0);
        __builtin_amdgcn_s_wait_tensorcnt(0);
    }
#else
    for (int idx = threadIdx.x; idx < 16 * CH; idx += 256) {
        const int r = idx >> 8, c = idx & (CH - 1);
        slab[r * SLAB_LD + c] = qg[(size_t)r * CH + c];
    }
#endif
    __syncthreads();

    const int m0 = (blockIdx.y * 8 + wave) * 16;
    const float* __restrict__ aRow = xg + (size_t)(m0 + l16) * CH;
    const float* __restrict__ bRow = slab + (size_t)l16 * SLAB_LD;

    v8f acc = {};
    #pragma unroll 8
    for (int k = 0; k < CH; k += 4) {
        const int kk = k + 2 * half;
        v2f a = *(const v2f*)(aRow + kk);   // global b64, x streamed once
        v2f b = *(const v2f*)(bRow + kk);   // ds b64, conflict-free
        acc = __builtin_amdgcn_wmma_f32_16x16x4_f32(
                  false, a, false, b, (short)0, acc, false, false);
    }

    #pragma unroll
    for (int r = 0; r < 8; ++r) {
        const int row = m0 + r + 8 * half;
        __builtin_nontemporal_store(acc[r], og + (size_t)row * CH + n0 + l16);
    }
}

// ---------------------------------------------------------------------------
// Pipeline (per group g, V = weight[g], columns are the reflectors):
//   G  = V^T V                 gram           (WMMA)
//   T  = larft(G)              WY factor      (barrier-free recurrence)
//   Wm = V @ T                                (WMMA)
//   Qt = I - V @ Wm^T          ( = (H_0...H_255)^T )                 (WMMA)
//   out = x @ Qt^T             big streaming GEMM (TDM + LDS + WMMA)
// Workspace: G, T, Wm, Qt = 4 x 4 MiB = 16 MiB.
// ---------------------------------------------------------------------------
extern "C" void kernel_launch(void* const* d_in, const int* in_sizes, int n_in,
                              void* d_out, int out_size, void* d_ws, size_t ws_size,
                              hipStream_t stream)
{
    const float* x = (const float*)d_in[0];   // (16, 8192, 256)
    const float* w = (const float*)d_in[1];   // (16, 256, 256)
    float* out     = (float*)d_out;           // (16, 8192, 256)

    const size_t mat = (size_t)GROUP * CH * CH;
    float* G  = (float*)d_ws;
    float* T  = G  + mat;
    float* Wm = T  + mat;
    float* Qt = Wm + mat;

    const long sM = (long)CH * CH;
    const long sX = (long)BS * CH;

    dim3 blk128(128);
    dim3 gridSq(CH / 16, CH / 16 / 4, GROUP);   // (16, 4, 16)

    // G = V^T V
    wmma_gemm_f32<true, false, false, false><<<gridSq, blk128, 0, stream>>>(
        w, w, G, CH, CH, CH, sM, sM, sM, CH, CH, CH);

    // T = larft(G)
    larft_f32<<<GROUP, CH, (size_t)CH * (CH + 1) * sizeof(float), stream>>>(G, T);

    // Wm = V @ T
    wmma_gemm_f32<false, false, false, false><<<gridSq, blk128, 0, stream>>>(
        w, T, Wm, CH, CH, CH, sM, sM, sM, CH, CH, CH);

    // Qt = I - V @ Wm^T   (transposed product matrix -> contiguous B frags later)
    wmma_gemm_f32<false, true, true, false><<<gridSq, blk128, 0, stream>>>(
        w, Wm, Qt, CH, CH, CH, sM, sM, sM, CH, CH, CH);

    // out = x @ Qt^T   (TDM-staged B slab, nontemporal streaming stores)
    dim3 gridX(CH / 16, BS / 128, GROUP);       // (16, 64, 16)
    xq_gemm_tdm<<<gridX, dim3(256), (size_t)(16 * SLAB_LD * 4), stream>>>(
        x, Qt, out);
}